// EGALayer_31542239822222
// MI455X (gfx1250) — compile-verified
//
#include <hip/hip_runtime.h>
#include <hip/hip_bf16.h>
#include <math.h>

typedef __attribute__((ext_vector_type(2))) float v2f;
typedef __attribute__((ext_vector_type(8))) float v8f;

#define B_ 4
#define N_ 512
#define D_ 128
#define H_ 8
#define DK_ 16
#define DFF_ 512
#define NBUCK_ 32

static __device__ __forceinline__ v8f wmma_f32(v2f a, v2f b, v8f c) {
    return __builtin_amdgcn_wmma_f32_16x16x4_f32(false, a, false, b, (short)0, c,
                                                 false, false);
}

// ---------------------------------------------------------------------------
// Generic fp32 WMMA GEMM: C[M,N] = A[M,K] @ W[K,N] (+bias)(+residual)(relu?)
// One wave computes one 16x16 tile. M,N,K multiples of 16. blockDim = 128.
// ---------------------------------------------------------------------------
__global__ __launch_bounds__(128) void ega_gemm_kernel(
    const float* __restrict__ A, const float* __restrict__ W,
    const float* __restrict__ bias, const float* __restrict__ residual,
    float* __restrict__ C, int M, int N, int K, int relu) {
    const int tilesN = N >> 4;
    const int tile = blockIdx.x * 4 + (threadIdx.x >> 5);
    const int tm = tile / tilesN;
    const int tn = tile % tilesN;
    const int lane = threadIdx.x & 31;
    const int lo = lane & 15;
    const int hi = lane >> 4;

    const int row = (tm << 4) + lo;   // A-fragment row for this lane
    const int col = (tn << 4) + lo;   // B-fragment column for this lane

    v8f acc = {};
    const float* Arow = A + (size_t)row * K;
#pragma unroll 4
    for (int k0 = 0; k0 < K; k0 += 4) {
        const int kb = k0 + (hi << 1);
        // gfx1250 speculative prefetch (global_prefetch_b8) one k-chunk ahead:
        // the strided W rows are the least cache-friendly stream here.
        __builtin_prefetch(W + (size_t)(kb + 4) * N + col, 0, 3);
        // A 16x4 f32: lanes 0-15 hold k0,k0+1; lanes 16-31 hold k0+2,k0+3
        float2 av = *(const float2*)(Arow + kb);
        v2f a; a.x = av.x; a.y = av.y;
        // B 4x16 f32: lane = column; same k split across lane halves
        v2f b; b.x = W[(size_t)kb * N + col]; b.y = W[(size_t)(kb + 1) * N + col];
        acc = wmma_f32(a, b, acc);
    }

    // C layout: VGPR j -> row (j + 8*hi), lanes 0..15 (mod 16) -> column
    const float bc = bias ? bias[col] : 0.0f;
#pragma unroll
    for (int j = 0; j < 8; ++j) {
        const int r = (tm << 4) + j + (hi << 3);
        float v = acc[j] + bc;
        if (residual) v += residual[(size_t)r * N + col];
        if (relu) v = fmaxf(v, 0.0f);
        C[(size_t)r * N + col] = v;
    }
}

// ---------------------------------------------------------------------------
// P[b,n,h,bucket] = dot(Q[b,n,h,:], emb[bucket, h*16 : h*16+16])
// ---------------------------------------------------------------------------
__global__ __launch_bounds__(256) void ega_pbucket_kernel(
    const float* __restrict__ Q, const float* __restrict__ emb,
    float* __restrict__ P) {
    const int idx = blockIdx.x * 256 + threadIdx.x;   // B*N*H*32 total
    const int bk = idx & 31;
    const int rest = idx >> 5;
    const int h = rest & 7;
    const int bn = rest >> 3;                          // 0 .. B*N-1
    const float* q = Q + (size_t)bn * D_ + h * DK_;
    const float* e = emb + (size_t)bk * D_ + h * DK_;
    float s = 0.0f;
#pragma unroll
    for (int d = 0; d < DK_; ++d) s += q[d] * e[d];
    P[idx] = s;
}

// ---------------------------------------------------------------------------
// Fused flash attention per (b, h, 16-row q-tile). One wave per block.
//   S = Q Kt / 4 + P[bucket(dist)]  -> online softmax -> O += prob @ V
// ---------------------------------------------------------------------------
__global__ __launch_bounds__(32) void ega_attn_kernel(
    const float* __restrict__ Q, const float* __restrict__ K,
    const float* __restrict__ V, const float* __restrict__ coords,
    const float* __restrict__ P, float* __restrict__ O) {
    const int bid = blockIdx.x;            // B*H*(N/16) = 1024
    const int qt = bid & 31;
    const int h = (bid >> 5) & 7;
    const int b = bid >> 8;
    const int lane = threadIdx.x;
    const int lo = lane & 15;
    const int hi = lane >> 4;

    __shared__ float sS[16 * 17];

    // Preload Q A-fragments for all 4 k-chunks (Dk = 16)
    const int n_row = (qt << 4) + lo;
    const float* Qp = Q + (((size_t)b * N_ + n_row) * H_ + h) * DK_;
    v2f qf[4];
#pragma unroll
    for (int c = 0; c < 4; ++c) {
        const int kb = 4 * c + (hi << 1);
        qf[c].x = Qp[kb];
        qf[c].y = Qp[kb + 1];
    }

    // Per-row (j) state: coords of row, P gather base, softmax running stats
    float crx[8], cry[8], mr[8], lr[8];
    int pb[8];
#pragma unroll
    for (int j = 0; j < 8; ++j) {
        const int n = (qt << 4) + j + (hi << 3);
        crx[j] = coords[((size_t)b * N_ + n) * 2 + 0];
        cry[j] = coords[((size_t)b * N_ + n) * 2 + 1];
        pb[j] = (((b * N_ + n) * H_) + h) * NBUCK_;
        mr[j] = -3.0e38f;
        lr[j] = 0.0f;
    }
    v8f o = {};

    for (int mt = 0; mt < (N_ >> 4); ++mt) {
        // ---- scores tile: S = Q @ K^T ----
        const int m_col = (mt << 4) + lo;
        const float* Kp = K + (((size_t)b * N_ + m_col) * H_ + h) * DK_;
        // prefetch next m-tile's K row for this lane (speculative, OOB-safe)
        __builtin_prefetch(Kp + 16 * H_ * DK_, 0, 3);
        v8f s = {};
#pragma unroll
        for (int c = 0; c < 4; ++c) {
            const int kb = 4 * c + (hi << 1);
            v2f kf; kf.x = Kp[kb]; kf.y = Kp[kb + 1];
            s = wmma_f32(qf[c], kf, s);
        }

        // column coords for this lane's column (fixed within the tile)
        const float2 cc = ((const float2*)coords)[(size_t)b * N_ + m_col];

        float p8[8];
#pragma unroll
        for (int j = 0; j < 8; ++j) {
            float sv = s[j] * 0.25f;                       // 1/sqrt(Dk)
            const float dx = crx[j] - cc.x;
            const float dy = cry[j] - cc.y;
            const float dist = sqrtf(dx * dx + dy * dy);
            int bk = (int)(dist * (float)NBUCK_);
            bk = bk < 0 ? 0 : (bk > NBUCK_ - 1 ? NBUCK_ - 1 : bk);
            sv += P[pb[j] + bk];

            // row-max over the 16 lanes holding this row
            float tmax = sv;
            tmax = fmaxf(tmax, __shfl_xor(tmax, 1, 32));
            tmax = fmaxf(tmax, __shfl_xor(tmax, 2, 32));
            tmax = fmaxf(tmax, __shfl_xor(tmax, 4, 32));
            tmax = fmaxf(tmax, __shfl_xor(tmax, 8, 32));

            const float mn = fmaxf(mr[j], tmax);
            const float al = __expf(mr[j] - mn);
            const float pe = __expf(sv - mn);
            float ts = pe;
            ts += __shfl_xor(ts, 1, 32);
            ts += __shfl_xor(ts, 2, 32);
            ts += __shfl_xor(ts, 4, 32);
            ts += __shfl_xor(ts, 8, 32);
            lr[j] = lr[j] * al + ts;
            o[j] *= al;
            mr[j] = mn;
            p8[j] = pe;
        }

        // ---- transpose probs (C layout) -> A layout via LDS ----
        __syncthreads();
#pragma unroll
        for (int j = 0; j < 8; ++j) sS[(j + (hi << 3)) * 17 + lo] = p8[j];
        __syncthreads();

        // ---- O += prob @ V ----
#pragma unroll
        for (int c = 0; c < 4; ++c) {
            const int kb = 4 * c + (hi << 1);
            v2f af; af.x = sS[lo * 17 + kb]; af.y = sS[lo * 17 + kb + 1];
            const int m0 = (mt << 4) + kb;
            v2f bf;
            bf.x = V[(((size_t)b * N_ + m0) * H_ + h) * DK_ + lo];
            bf.y = V[(((size_t)b * N_ + m0 + 1) * H_ + h) * DK_ + lo];
            o = wmma_f32(af, bf, o);
        }
    }

#pragma unroll
    for (int j = 0; j < 8; ++j) {
        const int n = (qt << 4) + j + (hi << 3);
        O[(((size_t)b * N_ + n) * H_ + h) * DK_ + lo] = o[j] / lr[j];
    }
}

// ---------------------------------------------------------------------------
// InstanceNorm over the N axis per (b, channel): one wave per (b,d) column.
// ---------------------------------------------------------------------------
__global__ __launch_bounds__(32) void ega_inorm_kernel(
    const float* __restrict__ x, const float* __restrict__ gamma,
    const float* __restrict__ beta, float* __restrict__ y) {
    const int b = blockIdx.x >> 7;   // D_ = 128
    const int d = blockIdx.x & 127;
    const int lane = threadIdx.x;

    float xs[16];
    float s = 0.0f, ss = 0.0f;
#pragma unroll
    for (int i = 0; i < 16; ++i) {
        const int n = lane + 32 * i;
        const float v = x[((size_t)b * N_ + n) * D_ + d];
        xs[i] = v;
        s += v;
        ss += v * v;
    }
#pragma unroll
    for (int off = 1; off < 32; off <<= 1) {
        s += __shfl_xor(s, off, 32);
        ss += __shfl_xor(ss, off, 32);
    }
    const float mean = s * (1.0f / (float)N_);
    const float var = ss * (1.0f / (float)N_) - mean * mean;   // biased
    const float r = rsqrtf(var + 1e-5f);
    const float rg = r * gamma[d];
    const float ofs = beta[d] - mean * rg;
#pragma unroll
    for (int i = 0; i < 16; ++i) {
        const int n = lane + 32 * i;
        y[((size_t)b * N_ + n) * D_ + d] = xs[i] * rg + ofs;
    }
}

// ---------------------------------------------------------------------------
extern "C" void kernel_launch(void* const* d_in, const int* in_sizes, int n_in,
                              void* d_out, int out_size, void* d_ws, size_t ws_size,
                              hipStream_t stream) {
    const float* h      = (const float*)d_in[0];
    const float* coords = (const float*)d_in[1];
    const float* W_Q    = (const float*)d_in[2];
    const float* W_K    = (const float*)d_in[3];
    const float* W_V    = (const float*)d_in[4];
    const float* W_O    = (const float*)d_in[5];
    const float* emb    = (const float*)d_in[6];
    const float* ffn_w1 = (const float*)d_in[7];
    const float* ffn_b1 = (const float*)d_in[8];
    const float* ffn_w2 = (const float*)d_in[9];
    const float* ffn_b2 = (const float*)d_in[10];
    const float* gamma1 = (const float*)d_in[11];
    const float* beta1  = (const float*)d_in[12];
    const float* gamma2 = (const float*)d_in[13];
    const float* beta2  = (const float*)d_in[14];
    float* out = (float*)d_out;

    const size_t BND = (size_t)B_ * N_ * D_;          // 262144
    float* ws = (float*)d_ws;
    float* Qb = ws;                 // [B,N,H,Dk]
    float* Kb = Qb + BND;
    float* Vb = Kb + BND;
    float* Pb = Vb + BND;           // [B,N,H,32]  (B*N*H*32 = 524288)
    float* Ob = Pb + (size_t)B_ * N_ * H_ * NBUCK_;
    float* t1 = Ob + BND;           // h + attn
    float* h1 = t1 + BND;           // norm1 output
    float* t2 = h1 + BND;           // FFN hidden [B*N, 512]
    float* t3 = t2 + (size_t)B_ * N_ * DFF_;

    const int M = B_ * N_;                              // 2048
    const int gemmDD = (M / 16) * (D_ / 16) / 4;        // 256 blocks
    const int gemmDF = (M / 16) * (DFF_ / 16) / 4;      // 1024 blocks

    // QKV projections
    ega_gemm_kernel<<<gemmDD, 128, 0, stream>>>(h, W_Q, nullptr, nullptr, Qb, M, D_, D_, 0);
    ega_gemm_kernel<<<gemmDD, 128, 0, stream>>>(h, W_K, nullptr, nullptr, Kb, M, D_, D_, 0);
    ega_gemm_kernel<<<gemmDD, 128, 0, stream>>>(h, W_V, nullptr, nullptr, Vb, M, D_, D_, 0);

    // Bucket projections P[b,n,h,bucket]
    ega_pbucket_kernel<<<(B_ * N_ * H_ * NBUCK_) / 256, 256, 0, stream>>>(Qb, emb, Pb);

    // Fused attention (scores + bucket bias + softmax + @V)
    ega_attn_kernel<<<B_ * H_ * (N_ / 16), 32, 0, stream>>>(Qb, Kb, Vb, coords, Pb, Ob);

    // Output projection + residual, then instance norm 1
    ega_gemm_kernel<<<gemmDD, 128, 0, stream>>>(Ob, W_O, nullptr, h, t1, M, D_, D_, 0);
    ega_inorm_kernel<<<B_ * D_, 32, 0, stream>>>(t1, gamma1, beta1, h1);

    // FFN
    ega_gemm_kernel<<<gemmDF, 128, 0, stream>>>(h1, ffn_w1, ffn_b1, nullptr, t2, M, DFF_, D_, 1);
    ega_gemm_kernel<<<gemmDD, 128, 0, stream>>>(t2, ffn_w2, ffn_b2, h1, t3, M, D_, DFF_, 0);

    // Instance norm 2 -> output
    ega_inorm_kernel<<<B_ * D_, 32, 0, stream>>>(t3, gamma2, beta2, out);

    (void)in_sizes; (void)n_in; (void)out_size; (void)ws_size;
}